// EdgeEncoding_19387482375234
// MI455X (gfx1250) — compile-verified
//
#include <hip/hip_runtime.h>
#include <hip/hip_bf16.h>

// ---------------------------------------------------------------------------
// EdgeEncoding for MI455X (gfx1250, wave32)
//   enc = edge_attr @ W.T + b           -> V_WMMA_F32_16X16X4_F32
//   out[h*N*N + pair] += coeff*enc[node,h] -> LDS-staged gather + run-length
//                                            segment accumulation (pair_idx is
//                                            non-decreasing) + f32 atomics
// ---------------------------------------------------------------------------

#define NN_ (768 * 768)
#define NNODES 768
#define NCH 32

typedef __attribute__((ext_vector_type(2))) float v2f;
typedef __attribute__((ext_vector_type(8))) float v8f;

// --------------------------- Kernel 1: enc GEMM ----------------------------
// enc(768x32) = edge_attr(768x4) x W^T(4x32) + b.  One 16x16 tile per wave.
// Grid: 96 blocks x 32 threads (48 row tiles x 2 col tiles).
__global__ __launch_bounds__(32) void enc_wmma_kernel(
    const float* __restrict__ EA,    // 768 x 4
    const float* __restrict__ W,     // 32 x 4
    const float* __restrict__ bias,  // 32
    float* __restrict__ enc)         // 768 x 32
{
  const int tile = blockIdx.x;
  const int row0 = (tile >> 1) << 4;   // row tile * 16
  const int c0   = (tile & 1) << 4;    // col tile * 16
  const int lane = threadIdx.x & 31;
  const int m    = lane & 15;          // M (A) or N (B/C/D) within tile
  const int kb   = (lane >> 4) << 1;   // K base: lanes 0-15 -> K0,1; 16-31 -> K2,3

  // A 16x4 f32: VGPR0 = {K=kb}, VGPR1 = {K=kb+1}, M = lane&15
  v2f a;
  a.x = EA[(row0 + m) * 4 + kb + 0];
  a.y = EA[(row0 + m) * 4 + kb + 1];

  // B 4x16 f32 (= W^T tile): B[k][n] = W[c0+n][k]; same K striping as A
  v2f bmat;
  bmat.x = W[(c0 + m) * 4 + kb + 0];
  bmat.y = W[(c0 + m) * 4 + kb + 1];

  // C 16x16 f32: VGPR r -> M=r (lanes 0-15) / M=r+8 (lanes 16-31), N = lane&15.
  // Bias is constant along M, so every accumulator reg = bias[c0 + n].
  const float bv = bias[c0 + m];
  v8f c;
#pragma unroll
  for (int r = 0; r < 8; ++r) c[r] = bv;

  // D = A x B + C   (8 args: neg_a, A, neg_b, B, c_mod, C, reuse_a, reuse_b)
  v8f d = __builtin_amdgcn_wmma_f32_16x16x4_f32(
      false, a, false, bmat, (short)0, c, false, false);

  // D layout mirrors C: VGPR r holds row row0 + r (+8 for upper lane half).
  const int mrow = row0 + ((lane >> 4) << 3);
#pragma unroll
  for (int r = 0; r < 8; ++r)
    enc[(mrow + r) * NCH + c0 + m] = d[r];
}

// --------------------------- Kernel 2: zero out ----------------------------
__global__ __launch_bounds__(256) void zero_kernel(float4* __restrict__ out, int n4)
{
  int i = blockIdx.x * blockDim.x + threadIdx.x;
  if (i < n4) out[i] = make_float4(0.f, 0.f, 0.f, 0.f);
}

// ----------------------- Kernel 3: segmented scatter -----------------------
// Wave-per-chunk, lane-per-channel. pair_idx is non-decreasing, so we
// accumulate per-segment in a register and flush with one f32 atomic per lane
// only when the segment id changes (avg segment ~17 entries => ~17x fewer
// atomics). Index triples are loaded 32-wide coalesced, then broadcast with
// v_readlane (uniform loop index -> scalar values, uniform branches).
__global__ __launch_bounds__(256) void segsum_kernel(
    const int*   __restrict__ pair_idx,
    const int*   __restrict__ node_idx,
    const float* __restrict__ coeff,
    const float* __restrict__ enc_g,   // 768 x 32 (from d_ws)
    float*       __restrict__ out,     // 32 x 768 x 768
    int nentries, int chunk)
{
  __shared__ float enc[NNODES * NCH];  // 96 KB of the 320 KB WGP LDS

  // Cooperative 128-bit LDS fill of the enc table.
  {
    const float4* src = (const float4*)enc_g;
    float4* dst = (float4*)enc;
    for (int i = threadIdx.x; i < (NNODES * NCH) / 4; i += blockDim.x)
      dst[i] = src[i];
  }
  __syncthreads();

  const int wave = (blockIdx.x * blockDim.x + threadIdx.x) >> 5;
  const int lane = threadIdx.x & 31;

  int begin = wave * chunk;
  int end   = begin + chunk;
  if (end > nentries) end = nentries;
  if (begin >= end) return;

  float* outLane = out + (size_t)lane * NN_;

  int   cur = -1;
  float acc = 0.f;

  for (int b = begin; b < end; b += 32) {
    int valid = end - b;
    if (valid > 32) valid = 32;

    int   p_l = 0, n_l = 0;
    float c_l = 0.f;
    if (lane < valid) {
      const int idx = b + lane;
      p_l = pair_idx[idx];
      n_l = node_idx[idx];
      c_l = coeff[idx];
    }

    for (int e = 0; e < valid; ++e) {
      const int   p  = __builtin_amdgcn_readlane(p_l, e);
      const int   nn = __builtin_amdgcn_readlane(n_l, e);
      const float cc = __uint_as_float(
          (unsigned)__builtin_amdgcn_readlane((int)__float_as_uint(c_l), e));

      if (p != cur) {                       // uniform branch across the wave
        if (cur >= 0) atomicAdd(&outLane[cur], acc);
        cur = p;
        acc = 0.f;
      }
      acc = fmaf(cc, enc[nn * NCH + lane], acc);  // conflict-free ds_load_b32
    }
  }
  if (cur >= 0) atomicAdd(&outLane[cur], acc);
}

// ------------------------------- Launcher ----------------------------------
extern "C" void kernel_launch(void* const* d_in, const int* in_sizes, int n_in,
                              void* d_out, int out_size, void* d_ws, size_t ws_size,
                              hipStream_t stream) {
  // setup_inputs order:
  // 0:x 1:edge_attr 2:W 3:b 4:edge_idx(i64) 5:pair_idx 6:node_idx 7:coeff 8:num_nodes
  const float* edge_attr = (const float*)d_in[1];
  const float* W         = (const float*)d_in[2];
  const float* bias      = (const float*)d_in[3];
  const int*   pair_idx  = (const int*)d_in[5];
  const int*   node_idx  = (const int*)d_in[6];
  const float* coeff     = (const float*)d_in[7];
  const int    nentries  = in_sizes[5];

  float* out = (float*)d_out;
  float* enc = (float*)d_ws;  // 768*32*4 = 96 KB scratch

  // 1) enc = edge_attr @ W.T + b via V_WMMA_F32_16X16X4_F32
  enc_wmma_kernel<<<96, 32, 0, stream>>>(edge_attr, W, bias, enc);

  // 2) zero the (32,768,768) output
  const int n4 = out_size / 4;
  zero_kernel<<<(n4 + 255) / 256, 256, 0, stream>>>((float4*)out, n4);

  // 3) segmented scatter-accumulate
  const int blocks = 512;                    // 512 blocks * 8 waves = 4096 waves
  const int waves  = blocks * 8;
  int chunk = (nentries + waves - 1) / waves;
  chunk = (chunk + 31) & ~31;                // batch-aligned chunks
  segsum_kernel<<<blocks, 256, 0, stream>>>(pair_idx, node_idx, coeff, enc, out,
                                            nentries, chunk);
}